// ASKLoss_70729521430619
// MI455X (gfx1250) — compile-verified
//
#include <hip/hip_runtime.h>
#include <hip/hip_bf16.h>

typedef __attribute__((ext_vector_type(2))) float v2f;
typedef __attribute__((ext_vector_type(8))) float v8f;

#define NREF   50000
#define BROWS  1024
#define DIMK   128
#define SLICES 25
#define CHUNKS 16      // chunks per slice
#define CW     128     // columns per chunk
#define XS     132     // LDS row stride (dwords): 132%64=4 -> conflict-free, 528B row -> 16B aligned

// ---------------- row-norm precompute: one wave per row, coalesced ----------------
__global__ __launch_bounds__(256) void askloss_norms(
    const float* __restrict__ x, const float* __restrict__ xref,
    float* __restrict__ xnorm, float* __restrict__ xrnorm)
{
    const int wid  = (blockIdx.x * blockDim.x + threadIdx.x) >> 5; // global wave id == row id
    const int lane = threadIdx.x & 31;
    const int R = BROWS + NREF;
    if (wid >= R) return;
    const float* row = (wid < BROWS) ? (x + (size_t)wid * DIMK)
                                     : (xref + (size_t)(wid - BROWS) * DIMK);
    float4 v = ((const float4*)row)[lane];            // 128 floats / 32 lanes = 1 float4 each
    float s = v.x * v.x + v.y * v.y + v.z * v.z + v.w * v.w;
    #pragma unroll
    for (int off = 16; off; off >>= 1) s += __shfl_xor(s, off, 32);
    if (lane == 0) {
        if (wid < BROWS) xnorm[wid] = s;
        else             xrnorm[wid - BROWS] = s;
    }
}

// ---------------- fused distance + exp + (Z, target-class) accumulation ----------------
__global__ __launch_bounds__(256) void askloss_main(
    const float* __restrict__ x, const float* __restrict__ xref,
    const int* __restrict__ y, const int* __restrict__ yref,
    const float* __restrict__ xnorm, const float* __restrict__ xrnorm,
    float* __restrict__ part)
{
    __shared__ float xlds[16 * XS];        // x tile 16 x 128 (padded)  ~8.4KB
    __shared__ float xrlds[128 * XS];      // xref chunk 128 x 128 (padded) ~67.6KB
    __shared__ float xrn[128];             // xref norms for chunk
    __shared__ float xxs[16];              // x norms for row tile
    __shared__ int   ycs[16];              // target classes for row tile
    __shared__ float wsum[8][16][2];       // per-wave per-row (Z,T)

    const int tid  = threadIdx.x;
    const int w    = tid >> 5;
    const int lane = tid & 31;
    const int lh   = lane >> 4;            // half-wave select
    const int ln   = lane & 15;
    const int b0   = blockIdx.x * 16;      // row-tile base
    const int sl   = blockIdx.y;           // N-slice

    // stage x tile (16 x 128) -> LDS
    #pragma unroll
    for (int i = 0; i < 2; ++i) {
        int flat = i * 256 + tid;                  // 512 float4 total
        int r = flat >> 5, c4 = flat & 31;
        float4 v = *(const float4*)&x[(size_t)(b0 + r) * DIMK + c4 * 4];
        *(float4*)&xlds[r * XS + c4 * 4] = v;
    }
    if (tid < 16) { xxs[tid] = xnorm[b0 + tid]; ycs[tid] = y[b0 + tid]; }
    __syncthreads();

    // A fragments are invariant across all chunks: keep them register-resident.
    // Lane (m + 16h) holds x[m][4i + 2h .. 4i + 2h + 1] for k-step i.
    v2f afrag[DIMK / 4];
    #pragma unroll
    for (int i = 0; i < DIMK / 4; ++i)
        afrag[i] = *(const v2f*)&xlds[ln * XS + i * 4 + 2 * lh];

    float xxv[8]; int ycv[8];
    #pragma unroll
    for (int j = 0; j < 8; ++j) { int m = j + 8 * lh; xxv[j] = xxs[m]; ycv[j] = ycs[m]; }

    float Zacc[8] = {0,0,0,0,0,0,0,0};
    float Tacc[8] = {0,0,0,0,0,0,0,0};

    for (int ci = 0; ci < CHUNKS; ++ci) {
        const int nbase = sl * (CHUNKS * CW) + ci * CW;
        if (nbase >= NREF) break;                  // uniform across block

        __syncthreads();                           // protect LDS from previous chunk readers
        // stage xref rows [nbase, nbase+128) x full K -> LDS (contiguous 64KB global read)
        #pragma unroll
        for (int i = 0; i < 16; ++i) {
            int flat = i * 256 + tid;              // 4096 float4 total
            int r = flat >> 5, c4 = flat & 31;
            int nr = nbase + r; nr = nr < NREF ? nr : NREF - 1;   // clamp; masked later
            float4 v = *(const float4*)&xref[(size_t)nr * DIMK + c4 * 4];
            *(float4*)&xrlds[r * XS + c4 * 4] = v;
        }
        if (tid < 128) {
            int nr = nbase + tid; nr = nr < NREF ? nr : NREF - 1;
            xrn[tid] = xrnorm[nr];
        }
        __syncthreads();

        // 16 x 16 tile, K=128 via 32 exact-f32 WMMAs (16x16x4); A from registers, B from LDS
        v8f acc = {0.f,0.f,0.f,0.f,0.f,0.f,0.f,0.f};
        #pragma unroll
        for (int i = 0; i < DIMK / 4; ++i) {
            // B frag: lane (n + 16h) holds xref[n][4i + 2h .. 4i + 2h + 1]
            v2f b = *(const v2f*)&xrlds[(w * 16 + ln) * XS + i * 4 + 2 * lh];
            acc = __builtin_amdgcn_wmma_f32_16x16x4_f32(
                    false, afrag[i], false, b, (short)0, acc, false, false);
        }

        // epilogue: scores -> exp -> additive (Z, T) accumulation (no max shift needed: score <= 0)
        const int  n     = nbase + w * 16 + ln;    // this lane's column
        const bool valid = (n < NREF);
        const int  nc    = valid ? n : NREF - 1;
        const float xr2  = xrn[w * 16 + ln];
        const int  cls   = yref[nc];
        #pragma unroll
        for (int j = 0; j < 8; ++j) {
            float d2 = xxv[j] - 2.0f * acc[j] + xr2;
            d2 = fmaxf(d2, 0.0f);
            float e = __expf(-sqrtf(d2));
            e = valid ? e : 0.0f;
            Zacc[j] += e;
            Tacc[j] += (cls == ycv[j]) ? e : 0.0f;
        }
    }

    // deterministic reduction: butterfly over the 16 lanes of each half-wave
    #pragma unroll
    for (int j = 0; j < 8; ++j) {
        #pragma unroll
        for (int off = 8; off; off >>= 1) {
            Zacc[j] += __shfl_xor(Zacc[j], off, 32);
            Tacc[j] += __shfl_xor(Tacc[j], off, 32);
        }
    }
    if (ln == 0) {
        #pragma unroll
        for (int j = 0; j < 8; ++j) {
            int m = j + 8 * lh;
            wsum[w][m][0] = Zacc[j];
            wsum[w][m][1] = Tacc[j];
        }
    }
    __syncthreads();
    if (tid < 16) {                                 // fixed-order cross-wave sum
        float Z = 0.f, T = 0.f;
        #pragma unroll
        for (int ww = 0; ww < 8; ++ww) { Z += wsum[ww][tid][0]; T += wsum[ww][tid][1]; }
        size_t o = ((size_t)sl * BROWS + b0 + tid) * 2;
        part[o]     = Z;
        part[o + 1] = T;
    }
}

// ---------------- final deterministic reduction over slices + rows ----------------
__global__ __launch_bounds__(1024) void askloss_finalize(
    const float* __restrict__ part, float* __restrict__ out)
{
    __shared__ float red[1024];
    const int b = threadIdx.x;
    float Z = 0.f, T = 0.f;
    for (int sl = 0; sl < SLICES; ++sl) {           // fixed order
        size_t o = ((size_t)sl * BROWS + b) * 2;
        Z += part[o];
        T += part[o + 1];
    }
    red[b] = -logf(T / Z + 1e-6f);
    __syncthreads();
    for (int s = 512; s > 0; s >>= 1) {
        if (b < s) red[b] += red[b + s];
        __syncthreads();
    }
    if (b == 0) out[0] = red[0] * (1.0f / (float)BROWS);
}

extern "C" void kernel_launch(void* const* d_in, const int* in_sizes, int n_in,
                              void* d_out, int out_size, void* d_ws, size_t ws_size,
                              hipStream_t stream) {
    const float* x    = (const float*)d_in[0];   // [1024,128]
    const float* xref = (const float*)d_in[1];   // [50000,128]
    const int*   y    = (const int*)d_in[2];     // [1024]
    const int*   yref = (const int*)d_in[3];     // [50000]
    float* out = (float*)d_out;

    float* ws     = (float*)d_ws;
    float* xnorm  = ws;                          // 1024 floats
    float* xrnorm = ws + BROWS;                  // 50000 floats
    float* part   = ws + BROWS + NREF;           // 25*1024*2 floats

    // norms: one wave per row, (1024+50000)/8 rows per block of 256
    const int nrm_blocks = (BROWS + NREF + 7) / 8;
    askloss_norms<<<nrm_blocks, 256, 0, stream>>>(x, xref, xnorm, xrnorm);

    askloss_main<<<dim3(BROWS / 16, SLICES), 256, 0, stream>>>(
        x, xref, y, yref, xnorm, xrnorm, part);

    askloss_finalize<<<1, 1024, 0, stream>>>(part, out);
}